// PathwayGNN_51573967291057
// MI455X (gfx1250) — compile-verified
//
#include <hip/hip_runtime.h>
#include <hip/hip_bf16.h>

#define NN 20000
#define EE 320000
#define EP (EE + NN)   // 340000 edges incl. self loops
#define GG 64
#define SLOPE 0.2f

typedef __attribute__((ext_vector_type(2))) float v2f;
typedef __attribute__((ext_vector_type(8))) float v8f;

// ---- order-preserving float<->uint mapping for atomic max (identity = 0) ----
__device__ __forceinline__ unsigned f2ord(float f) {
  unsigned b = __float_as_uint(f);
  return (b & 0x80000000u) ? ~b : (b | 0x80000000u);
}
__device__ __forceinline__ float ord2f(unsigned k) {
  unsigned b = (k & 0x80000000u) ? (k & 0x7FFFFFFFu) : ~k;
  return __uint_as_float(b);
}
__device__ __forceinline__ float lrelu(float x) { return x > 0.f ? x : SLOPE * x; }

// ---------------- edge_attr column sums (for self-loop fill) ----------------
__global__ void k_easum(const float* __restrict__ ea, float* __restrict__ easum) {
  __shared__ float s0[256], s1[256];
  int tid = threadIdx.x;
  float a0 = 0.f, a1 = 0.f;
  for (int e = blockIdx.x * blockDim.x + tid; e < EE; e += gridDim.x * blockDim.x) {
    a0 += ea[e * 2 + 0];
    a1 += ea[e * 2 + 1];
  }
  s0[tid] = a0; s1[tid] = a1;
  __syncthreads();
  for (int s = 128; s > 0; s >>= 1) {
    if (tid < s) { s0[tid] += s0[tid + s]; s1[tid] += s1[tid + s]; }
    __syncthreads();
  }
  if (tid == 0) { atomicAdd(&easum[0], s0[0]); atomicAdd(&easum[1], s1[0]); }
}

// ---- fold We @ a_edge  (ve[0..3]=layer1 h*2+k, ve[4..5]=layer2 k, ve[6..7]=ea mean) ----
__global__ void k_prep(const float* __restrict__ We1, const float* __restrict__ ae1,
                       const float* __restrict__ We2, const float* __restrict__ ae2,
                       const float* __restrict__ easum, float* __restrict__ ve) {
  if (threadIdx.x == 0 && blockIdx.x == 0) {
    for (int h = 0; h < 2; ++h)
      for (int k = 0; k < 2; ++k) {
        float s = 0.f;
        for (int c = 0; c < 16; ++c) s += We1[k * 32 + h * 16 + c] * ae1[h * 16 + c];
        ve[h * 2 + k] = s;
      }
    for (int k = 0; k < 2; ++k) {
      float s = 0.f;
      for (int c = 0; c < 256; ++c) s += We2[k * 256 + c] * ae2[c];
      ve[4 + k] = s;
    }
    ve[6] = easum[0] * (1.0f / EE);
    ve[7] = easum[1] * (1.0f / EE);
  }
}

// ---------------- layer1 GEMM: h1[N,32] = x[N,4] @ W1[4,32] (WMMA f32 K=4) ----------------
__global__ void k_gemm1(const float* __restrict__ x, const float* __restrict__ W1,
                        float* __restrict__ h1) {
  const int tile = blockIdx.x;          // one wave per 16-row tile (N % 16 == 0)
  const int lane = threadIdx.x & 31;
  const int m = lane & 15;
  const int row = tile * 16 + m;
  const int khalf = (lane >> 4) * 2;    // lanes 0-15 hold K=0,1; lanes 16-31 hold K=2,3
  v2f a;
  a.x = x[row * 4 + khalf + 0];
  a.y = x[row * 4 + khalf + 1];
  const int rbase = tile * 16 + ((lane >> 4) ? 8 : 0);
#pragma unroll
  for (int t = 0; t < 2; ++t) {
    const int n = t * 16 + m;
    v2f b;
    b.x = W1[(khalf + 0) * 32 + n];
    b.y = W1[(khalf + 1) * 32 + n];
    v8f c = {};
    c = __builtin_amdgcn_wmma_f32_16x16x4_f32(false, a, false, b, (short)0, c, false, false);
#pragma unroll
    for (int r = 0; r < 8; ++r) h1[(rbase + r) * 32 + n] = c[r];
  }
}

// ---------------- per-node attention coefficients, layer 1 ----------------
__global__ void k_coef1(const float* __restrict__ h1, const float* __restrict__ as1,
                        const float* __restrict__ ad1, float* __restrict__ asv,
                        float* __restrict__ adv) {
  int nno = blockIdx.x * blockDim.x + threadIdx.x;
  if (nno >= NN) return;
  for (int h = 0; h < 2; ++h) {
    float ss = 0.f, sd = 0.f;
    for (int c = 0; c < 16; ++c) {
      float v = h1[nno * 32 + h * 16 + c];
      ss += v * as1[h * 16 + c];
      sd += v * ad1[h * 16 + c];
    }
    asv[nno * 2 + h] = ss;
    adv[nno * 2 + h] = sd;
  }
}

// ---------------- layer1 alpha + segment max (real edges and self loops) ----------------
__global__ void k_alpha1(const int* __restrict__ ei, const float* __restrict__ ea,
                         const float* __restrict__ ve, const float* __restrict__ asv,
                         const float* __restrict__ adv, float* __restrict__ alpha1,
                         unsigned* __restrict__ amax1) {
  int e = blockIdx.x * blockDim.x + threadIdx.x;
  if (e >= EP) return;
  int s, d; float e0, e1;
  if (e < EE) { s = ei[e]; d = ei[EE + e]; e0 = ea[e * 2]; e1 = ea[e * 2 + 1]; }
  else        { s = d = e - EE; e0 = ve[6]; e1 = ve[7]; }
  for (int h = 0; h < 2; ++h) {
    float al = asv[s * 2 + h] + adv[d * 2 + h] + e0 * ve[h * 2 + 0] + e1 * ve[h * 2 + 1];
    al = lrelu(al);
    alpha1[e * 2 + h] = al;
    atomicMax(&amax1[d * 2 + h], f2ord(al));
  }
}

__global__ void k_den1(const int* __restrict__ ei, const float* __restrict__ alpha1,
                       const unsigned* __restrict__ amax1, float* __restrict__ den1) {
  int e = blockIdx.x * blockDim.x + threadIdx.x;
  if (e >= EP) return;
  int d = (e < EE) ? ei[EE + e] : (e - EE);
  for (int h = 0; h < 2; ++h) {
    float m = ord2f(amax1[d * 2 + h]);
    atomicAdd(&den1[d * 2 + h], __expf(alpha1[e * 2 + h] - m));
  }
}

// one wave per edge, 32 lanes = 32 features (H*C = 32)
__global__ void k_msg1(const int* __restrict__ ei, const float* __restrict__ alpha1,
                       const unsigned* __restrict__ amax1, const float* __restrict__ den1,
                       const float* __restrict__ h1, float* __restrict__ out1) {
  long idx = (long)blockIdx.x * blockDim.x + threadIdx.x;
  int e = (int)(idx >> 5);
  int f = (int)(idx & 31);
  if (e >= EP) return;
  int s, d;
  if (e < EE) { s = ei[e]; d = ei[EE + e]; } else { s = d = e - EE; }
  int h = f >> 4;
  float m = ord2f(amax1[d * 2 + h]);
  float w = __expf(alpha1[e * 2 + h] - m) / (den1[d * 2 + h] + 1e-16f);
  atomicAdd(&out1[d * 32 + f], h1[s * 32 + f] * w);
}

__global__ void k_finish1(float* __restrict__ out1, const float* __restrict__ b1) {
  int i = blockIdx.x * blockDim.x + threadIdx.x;
  if (i >= NN * 32) return;
  float v = out1[i] + b1[i & 31];
  out1[i] = v > 0.f ? v : 0.f;
}

// ---------------- layer2 GEMM: h2[N,256] = h1r[N,32] @ W2[32,256] (WMMA f32 K=4 x8) ----------------
__global__ void k_gemm2(const float* __restrict__ h, const float* __restrict__ W2,
                        float* __restrict__ h2) {
  int wave = blockIdx.x * (blockDim.x >> 5) + (threadIdx.x >> 5);
  if (wave >= NN / 16) return;        // wave-uniform: EXEC stays all-ones for WMMA
  const int lane = threadIdx.x & 31;
  const int m = lane & 15;
  const int row = wave * 16 + m;
  const int ksel = (lane >> 4) * 2;
  v2f a[8];
#pragma unroll
  for (int kk = 0; kk < 8; ++kk) {
    a[kk].x = h[row * 32 + kk * 4 + ksel + 0];
    a[kk].y = h[row * 32 + kk * 4 + ksel + 1];
  }
  const int rbase = wave * 16 + ((lane >> 4) ? 8 : 0);
  for (int t = 0; t < 16; ++t) {
    const int n = t * 16 + m;
    v8f c = {};
#pragma unroll
    for (int kk = 0; kk < 8; ++kk) {
      v2f b;
      b.x = W2[(kk * 4 + ksel + 0) * 256 + n];
      b.y = W2[(kk * 4 + ksel + 1) * 256 + n];
      c = __builtin_amdgcn_wmma_f32_16x16x4_f32(false, a[kk], false, b, (short)0, c, false, false);
    }
#pragma unroll
    for (int r = 0; r < 8; ++r) h2[(rbase + r) * 256 + n] = c[r];
  }
}

// ---------------- per-node coefficients, layer 2 (wave per node, shuffle reduce) ----------------
__global__ void k_coef2(const float* __restrict__ h2, const float* __restrict__ as2,
                        const float* __restrict__ ad2, float* __restrict__ asv,
                        float* __restrict__ adv) {
  int wave = blockIdx.x * (blockDim.x >> 5) + (threadIdx.x >> 5);
  int lane = threadIdx.x & 31;
  if (wave >= NN) return;
  float ss = 0.f, sd = 0.f;
  for (int c = lane; c < 256; c += 32) {
    float v = h2[wave * 256 + c];
    ss += v * as2[c];
    sd += v * ad2[c];
  }
  for (int o = 16; o > 0; o >>= 1) {
    ss += __shfl_down(ss, o, 32);
    sd += __shfl_down(sd, o, 32);
  }
  if (lane == 0) { asv[wave] = ss; adv[wave] = sd; }
}

__global__ void k_alpha2(const int* __restrict__ ei, const float* __restrict__ ea,
                         const float* __restrict__ ve, const float* __restrict__ asv,
                         const float* __restrict__ adv, float* __restrict__ alpha2,
                         unsigned* __restrict__ amax2) {
  int e = blockIdx.x * blockDim.x + threadIdx.x;
  if (e >= EP) return;
  int s, d; float e0, e1;
  if (e < EE) { s = ei[e]; d = ei[EE + e]; e0 = ea[e * 2]; e1 = ea[e * 2 + 1]; }
  else        { s = d = e - EE; e0 = ve[6]; e1 = ve[7]; }
  float al = asv[s] + adv[d] + e0 * ve[4] + e1 * ve[5];
  al = lrelu(al);
  alpha2[e] = al;
  atomicMax(&amax2[d], f2ord(al));
}

__global__ void k_den2(const int* __restrict__ ei, const float* __restrict__ alpha2,
                       const unsigned* __restrict__ amax2, float* __restrict__ den2) {
  int e = blockIdx.x * blockDim.x + threadIdx.x;
  if (e >= EP) return;
  int d = (e < EE) ? ei[EE + e] : (e - EE);
  atomicAdd(&den2[d], __expf(alpha2[e] - ord2f(amax2[d])));
}

// one 256-thread block per edge; atomics resolve in L2 (h2/out2 both fit in 192MB L2)
__global__ void k_msg2(const int* __restrict__ ei, const float* __restrict__ alpha2,
                       const unsigned* __restrict__ amax2, const float* __restrict__ den2,
                       const float* __restrict__ h2, float* __restrict__ out2) {
  int e = blockIdx.x;
  int c = threadIdx.x;
  int s, d;
  if (e < EE) { s = ei[e]; d = ei[EE + e]; } else { s = d = e - EE; }
  float w = __expf(alpha2[e] - ord2f(amax2[d])) / (den2[d] + 1e-16f);
  atomicAdd(&out2[d * 256 + c], h2[s * 256 + c] * w);
}

// bias + pooled sum into d_out, plus per-graph counts
__global__ void k_pool(const float* __restrict__ out2, const float* __restrict__ b2,
                       const int* __restrict__ batch, float* __restrict__ dout,
                       float* __restrict__ cnt) {
  int n = blockIdx.x;
  int c = threadIdx.x;
  int g = batch[n];
  atomicAdd(&dout[g * 256 + c], out2[n * 256 + c] + b2[c]);
  if (c == 0) atomicAdd(&cnt[g], 1.0f);
}

__global__ void k_pooldiv(float* __restrict__ dout, const float* __restrict__ cnt) {
  int i = blockIdx.x * blockDim.x + threadIdx.x;
  if (i >= GG * 256) return;
  dout[i] /= fmaxf(cnt[i >> 8], 1.0f);
}

extern "C" void kernel_launch(void* const* d_in, const int* in_sizes, int n_in,
                              void* d_out, int out_size, void* d_ws, size_t ws_size,
                              hipStream_t stream) {
  const float* x    = (const float*)d_in[0];
  const int*   ei   = (const int*)d_in[1];
  const float* ea   = (const float*)d_in[2];
  const int*   bat  = (const int*)d_in[3];
  const float* W1   = (const float*)d_in[4];
  const float* We1  = (const float*)d_in[5];
  const float* as1  = (const float*)d_in[6];
  const float* ad1  = (const float*)d_in[7];
  const float* ae1  = (const float*)d_in[8];
  const float* b1   = (const float*)d_in[9];
  const float* W2   = (const float*)d_in[10];
  const float* We2  = (const float*)d_in[11];
  const float* as2  = (const float*)d_in[12];
  const float* ad2  = (const float*)d_in[13];
  const float* ae2  = (const float*)d_in[14];
  const float* b2   = (const float*)d_in[15];
  float* dout = (float*)d_out;

  // -------- workspace layout (floats) --------
  float* w = (float*)d_ws;
  size_t o = 0;
  float*    h1     = w + o; o += (size_t)NN * 32;
  float*    asv1   = w + o; o += (size_t)NN * 2;
  float*    adv1   = w + o; o += (size_t)NN * 2;
  unsigned* amax1  = (unsigned*)(w + o); o += (size_t)NN * 2;
  float*    den1   = w + o; o += (size_t)NN * 2;
  float*    alpha1 = w + o; o += (size_t)EP * 2;
  float*    out1   = w + o; o += (size_t)NN * 32;
  float*    h2     = w + o; o += (size_t)NN * 256;
  float*    asv2   = w + o; o += (size_t)NN;
  float*    adv2   = w + o; o += (size_t)NN;
  unsigned* amax2  = (unsigned*)(w + o); o += (size_t)NN;
  float*    den2   = w + o; o += (size_t)NN;
  float*    alpha2 = w + o; o += (size_t)EP;
  float*    out2   = w + o; o += (size_t)NN * 256;
  float*    easum  = w + o; o += 2;
  float*    ve     = w + o; o += 8;
  float*    cnt    = w + o; o += GG;
  const size_t ws_used_bytes = o * sizeof(float);

  hipMemsetAsync(d_ws, 0, ws_used_bytes, stream);
  hipMemsetAsync(d_out, 0, (size_t)out_size * sizeof(float), stream);

  // edge-attr mean + folded edge-attention vectors
  k_easum<<<256, 256, 0, stream>>>(ea, easum);
  k_prep<<<1, 64, 0, stream>>>(We1, ae1, We2, ae2, easum, ve);

  // ---- layer 1 ----
  k_gemm1<<<NN / 16, 32, 0, stream>>>(x, W1, h1);
  k_coef1<<<(NN + 255) / 256, 256, 0, stream>>>(h1, as1, ad1, asv1, adv1);
  k_alpha1<<<(EP + 255) / 256, 256, 0, stream>>>(ei, ea, ve, asv1, adv1, alpha1, amax1);
  k_den1<<<(EP + 255) / 256, 256, 0, stream>>>(ei, alpha1, amax1, den1);
  k_msg1<<<(int)(((long)EP * 32 + 255) / 256), 256, 0, stream>>>(ei, alpha1, amax1, den1, h1, out1);
  k_finish1<<<(NN * 32 + 255) / 256, 256, 0, stream>>>(out1, b1);

  // ---- layer 2 ----
  k_gemm2<<<(NN / 16 + 3) / 4, 128, 0, stream>>>(out1, W2, h2);
  k_coef2<<<(NN * 32 + 255) / 256, 256, 0, stream>>>(h2, as2, ad2, asv2, adv2);
  k_alpha2<<<(EP + 255) / 256, 256, 0, stream>>>(ei, ea, ve, asv2, adv2, alpha2, amax2);
  k_den2<<<(EP + 255) / 256, 256, 0, stream>>>(ei, alpha2, amax2, den2);
  k_msg2<<<EP, 256, 0, stream>>>(ei, alpha2, amax2, den2, h2, out2);

  // ---- pool ----
  k_pool<<<NN, 256, 0, stream>>>(out2, b2, bat, dout, cnt);
  k_pooldiv<<<(GG * 256 + 255) / 256, 256, 0, stream>>>(dout, cnt);
}